// Quantizer_af4_14224931684493
// MI455X (gfx1250) — compile-verified
//
#include <hip/hip_runtime.h>
#include <stdint.h>

#define BLOCK 256

// Branchless nearest-code for the fixed AF4 codebook
// [-6,-4,-3,-2,-1.5,-1,-0.5,-0, 0,0.5,1,1.5,2,3,4,6].
// Works in the doubled domain a2 = 2*|x/scale|:
//   lower region codes {0,.5,1,1.5,2}  -> r = rndne(min(a2,4))   (r = 2*code)
//   upper region codes {3,4,6}         -> r = 6/8/12 via doubled thresholds 5/7/10
// Row-uniform precomputes: rs2p=2/sp, rs2n=-2/sn (so a2>=0), shp=sp/2, shn=-sn/2.
// out = r * (±scale/2).  ~14 VALU ops per element.
__device__ __forceinline__ float af4_one(float xv, float xmin, float xmax,
                                         float rs2p, float rs2n,
                                         float shp, float shn) {
    const float xc  = __builtin_amdgcn_fmed3f(xv, xmin, xmax);  // clamp, 1 op
    const bool  pos = (xc >= 0.0f);
    const float a2  = xc * (pos ? rs2p : rs2n);                 // 2|q| >= 0
    const float ssh = pos ? shp : shn;                          // ±scale/2
    float r = __builtin_rintf(fminf(a2, 4.0f));                 // v_rndne
    r = (a2 > 5.0f)  ? 6.0f  : r;
    r = (a2 > 7.0f)  ? 8.0f  : r;
    r = (a2 > 10.0f) ? 12.0f : r;
    return r * ssh;
}

// Fast path: one 256-thread block (8 wave32) per row. All ITERS tiles are
// staged into LDS with async b128 copies up-front (ASYNCcnt path), then
// drained with decreasing s_wait_asynccnt immediates. Each wave stages the
// exact LDS slots it consumes -> no workgroup barriers needed.
template <int ITERS>
__global__ __launch_bounds__(BLOCK)
void af4_quant_rows(const float* __restrict__ x,
                    const float* __restrict__ scale_pos,
                    const float* __restrict__ scale_neg,
                    const float* __restrict__ tmaxp,
                    const float* __restrict__ tminp,
                    float* __restrict__ out) {
    __shared__ float4 smem[ITERS][BLOCK];       // 4 KB per tile

    const int row = blockIdx.x;
    const int tid = threadIdx.x;
    const int cols = ITERS * BLOCK * 4;

    // Row-uniform parameters -> scalar loads; v_rcp (1 ulp) instead of IEEE div.
    const float sp   = scale_pos[row];
    const float sn   = scale_neg[row];
    const float xmax = tmaxp[0];
    const float xmin = tminp[0];
    const float rs2p =  2.0f * __builtin_amdgcn_rcpf(sp);
    const float rs2n = -2.0f * __builtin_amdgcn_rcpf(sn);
    const float shp  =  0.5f * sp;
    const float shn  = -0.5f * sn;

    const float* rowp = x + (size_t)row * (size_t)cols;
    float4*      orow = (float4*)(out + (size_t)row * (size_t)cols);

    // Wave-relative LDS byte offset (low 32 bits of the generic address).
    const uint32_t lds0 = (uint32_t)(uintptr_t)(&smem[0][0]);

    // Stage the whole row: ITERS async b128 copies in flight per lane-slot.
#pragma unroll
    for (int j = 0; j < ITERS; ++j) {
        uint64_t ga = (uint64_t)(uintptr_t)(rowp + (size_t)(j * BLOCK + tid) * 4u);
        uint32_t la = lds0 + (uint32_t)(j * BLOCK + tid) * 16u;
        asm volatile("global_load_async_to_lds_b128 %0, %1, off"
                     :: "v"(la), "v"(ga) : "memory");
    }

    // Drain: wait for tile j (in-order completion), compute, store b128.
#pragma unroll
    for (int j = 0; j < ITERS; ++j) {
        asm volatile("s_wait_asynccnt %0" :: "i"(ITERS - 1 - j) : "memory");
        float4 v = smem[j][tid];                // ds_load_b128
        float4 r;
        r.x = af4_one(v.x, xmin, xmax, rs2p, rs2n, shp, shn);
        r.y = af4_one(v.y, xmin, xmax, rs2p, rs2n, shp, shn);
        r.z = af4_one(v.z, xmin, xmax, rs2p, rs2n, shp, shn);
        r.w = af4_one(v.w, xmin, xmax, rs2p, rs2n, shp, shn);
        orow[(size_t)j * BLOCK + tid] = r;      // global_store_b128
    }
}

// Generic fallback for arbitrary shapes (scalar, grid-stride free).
__global__ void af4_quant_generic(const float* __restrict__ x,
                                  const float* __restrict__ scale_pos,
                                  const float* __restrict__ scale_neg,
                                  const float* __restrict__ tmaxp,
                                  const float* __restrict__ tminp,
                                  float* __restrict__ out,
                                  int n, int cols) {
    int i = blockIdx.x * blockDim.x + threadIdx.x;
    if (i >= n) return;
    int row = i / cols;
    float sp = scale_pos[row];
    float sn = scale_neg[row];
    out[i] = af4_one(x[i], tminp[0], tmaxp[0],
                     2.0f * __builtin_amdgcn_rcpf(sp),
                     -2.0f * __builtin_amdgcn_rcpf(sn),
                     0.5f * sp, -0.5f * sn);
}

extern "C" void kernel_launch(void* const* d_in, const int* in_sizes, int n_in,
                              void* d_out, int out_size, void* d_ws, size_t ws_size,
                              hipStream_t stream) {
    const float* x    = (const float*)d_in[0];  // [R, C]
    const float* sp   = (const float*)d_in[1];  // [R, 1]
    const float* sn   = (const float*)d_in[2];  // [R, 1]
    // d_in[3] = code: fixed AF4 codebook, thresholds baked into the kernel
    const float* tmax = (const float*)d_in[4];  // [1]
    const float* tmin = (const float*)d_in[5];  // [1]
    float* out = (float*)d_out;

    const int total = in_sizes[0];
    const int rows  = in_sizes[1];
    const int cols  = total / rows;

    switch (cols) {
    case 1 * BLOCK * 4:
        af4_quant_rows<1><<<rows, BLOCK, 0, stream>>>(x, sp, sn, tmax, tmin, out);
        break;
    case 2 * BLOCK * 4:
        af4_quant_rows<2><<<rows, BLOCK, 0, stream>>>(x, sp, sn, tmax, tmin, out);
        break;
    case 4 * BLOCK * 4:   // C = 4096: the reference shape
        af4_quant_rows<4><<<rows, BLOCK, 0, stream>>>(x, sp, sn, tmax, tmin, out);
        break;
    case 8 * BLOCK * 4:
        af4_quant_rows<8><<<rows, BLOCK, 0, stream>>>(x, sp, sn, tmax, tmin, out);
        break;
    default:
        af4_quant_generic<<<(total + BLOCK - 1) / BLOCK, BLOCK, 0, stream>>>(
            x, sp, sn, tmax, tmin, out, total, cols);
        break;
    }
}